// OnePoleDecoder_87823491268988
// MI455X (gfx1250) — compile-verified
//
#include <hip/hip_runtime.h>
#include <hip/hip_bf16.h>

typedef __attribute__((ext_vector_type(2))) float v2f;
typedef __attribute__((ext_vector_type(8))) float v8f;

#define T_LEN 32768
#define DIN   512
#define DH    2048
#define JB    64     // hidden columns staged per j-block (4 WMMA j-tiles)
#define KCH   128    // k-depth per LDS chunk (32 KB); double-buffered

// D = A(16x4 f32) * B(4x16 f32) + C(16x16 f32), full fp32 precision.
__device__ __forceinline__ v8f wmma_f32(v2f a, v2f b, v8f c) {
  return __builtin_amdgcn_wmma_f32_16x16x4_f32(
      /*neg_a=*/false, a, /*neg_b=*/false, b,
      /*c_mod=*/(short)0, c, /*reuse_a=*/false, /*reuse_b=*/false);
}

// Branch-free tanh: tanh(x) = (e^{2x}-1)/(e^{2x}+1); clamp keeps e finite.
__device__ __forceinline__ float fast_tanh(float x) {
  float xc = fminf(fmaxf(x, -10.0f), 10.0f);
  float e  = __expf(2.0f * xc);
  return (e - 1.0f) * __builtin_amdgcn_rcpf(e + 1.0f);
}

// Fused: X = tanh(beliefs @ W_in^T + b_in); s_theta = X @ w_theta; s_z = X @ w_z.
// 8 waves/WG, one 16-row t-tile per wave. W_in is shared across the WG through
// LDS, copied with async global->LDS b64 ops that scatter directly into the
// WMMA B-fragment layout, double-buffered so copy(c+1) overlaps compute(c).
__global__ void __launch_bounds__(256) onepole_gemm(
    const float* __restrict__ beliefs,  // [T, DIN]
    const float* __restrict__ W_in,     // [DH, DIN]
    const float* __restrict__ b_in,     // [DH]
    const float* __restrict__ W_theta,  // [DH]
    const float* __restrict__ W_z,      // [DH]
    float* __restrict__ s_theta,        // [T]
    float* __restrict__ s_z)            // [T]
{
  __shared__ float lds[2][JB * KCH];    // 2 x 32 KB, B-fragment layout

  const int tid  = (int)threadIdx.x;
  const int lane = tid & 31;
  const int wv   = tid >> 5;
  const int t0   = (int)blockIdx.x * 128 + wv * 16;
  const int m    = lane & 15;   // A row / B col within a 16-tile
  const int kh   = lane >> 4;   // k-half selector (K = 2*kh + {0,1})

  const float* Abase = beliefs + (size_t)(t0 + m) * DIN + 2 * kh;

  // ---- Per-thread staging decode: pair index p = 4*i + (lane&3), j row from
  // lane[4:2] | wv. k = 2p -> fragment (q = p>>1, h = p&1). Global side: 4
  // lanes x 8 B contiguous per row; LDS side: <=2-way bank groups.
  const int pl   = lane & 3;
  const int js   = (wv << 3) | ((lane >> 2) & 7);   // 0..63
  const int jt_s = js >> 4, n_s = js & 15;
  const int hh   = lane & 1;                         // h = p & 1
  const int q0   = (lane >> 1) & 1;                  // q low bit
  const unsigned fragoff =
      (unsigned)((((q0 * 4 + jt_s) * 32) + hh * 16 + n_s) * 8);
  const float* grow = W_in + (size_t)js * DIN + 2 * pl;

  // Issue one 32 KB chunk (64 j x 128 k) as 16 async b64 copies per thread,
  // scattered into the WMMA fragment layout. Tracked by ASYNCcnt.
  auto issue = [&](int jb, int kc, int b) {
    const float* gp = grow + (size_t)jb * DIN + kc;
    unsigned ldso = (unsigned)(size_t)(&lds[b][0]) + fragoff;
#pragma unroll
    for (int i = 0; i < 16; ++i) {
      asm volatile("global_load_async_to_lds_b64 %0, %1, off"
                   :: "v"(ldso), "v"(gp)
                   : "memory");
      ldso += 2048u;   // q advances by 2 per i
      gp   += 8;       // k advances by 8 per i
    }
  };

  float accth[8], accz[8];
#pragma unroll
  for (int r = 0; r < 8; ++r) { accth[r] = 0.0f; accz[r] = 0.0f; }

  issue(0, 0, 0);     // prologue: chunk 0 in flight
  int buf = 0;

  for (int jb = 0; jb < DH; jb += JB) {
    v8f c0 = {}; v8f c1 = {}; v8f c2 = {}; v8f c3 = {};

    for (int kc = 0; kc < DIN; kc += KCH) {
      // Kick off the next chunk into the other buffer, then wait for the
      // current chunk (<=16 outstanding leaves only the new issues pending).
      int njb = jb, nkc = kc + KCH;
      if (nkc == DIN) { nkc = 0; njb += JB; }
      if (njb < DH) {
        issue(njb, nkc, buf ^ 1);
        asm volatile("s_wait_asynccnt 0x10" ::: "memory");
      } else {
        asm volatile("s_wait_asynccnt 0x0" ::: "memory");
      }
      __syncthreads();  // all waves' copies for this chunk have landed

      // ---- Consume: A from global (b64), B from LDS (conflict-free b64).
      const float* Ak = Abase + kc;
      const v2f* bbuf = (const v2f*)(&lds[buf][0]);
#pragma unroll 4
      for (int q = 0; q < KCH / 4; ++q) {
        v2f a  = *(const v2f*)(Ak + 4 * q);
        const v2f* bp = bbuf + (size_t)q * 128 + lane;
        v2f b0 = bp[0];
        v2f b1 = bp[32];
        v2f b2 = bp[64];
        v2f b3 = bp[96];
        c0 = wmma_f32(a, b0, c0);
        c1 = wmma_f32(a, b1, c1);
        c2 = wmma_f32(a, b2, c2);
        c3 = wmma_f32(a, b3, c3);
      }
      __syncthreads();  // consumers done before this buffer is overwritten
      buf ^= 1;
    }

    // ---- Epilogue: C element (vgpr r, lane l) = X[t0+r+8*kh][jb+16*jt+m]
#pragma unroll
    for (int jt = 0; jt < 4; ++jt) {
      const int j = jb + 16 * jt + m;
      const float bi = b_in[j];
      const float wt = W_theta[j];
      const float wz = W_z[j];
      v8f c = (jt == 0) ? c0 : (jt == 1) ? c1 : (jt == 2) ? c2 : c3;
#pragma unroll
      for (int r = 0; r < 8; ++r) {
        float x = fast_tanh(c[r] + bi);
        accth[r] = fmaf(x, wt, accth[r]);
        accz[r]  = fmaf(x, wz, accz[r]);
      }
    }
  }

  // Reduce across the 16 lanes of each half-wave (cols n = 0..15).
#pragma unroll
  for (int r = 0; r < 8; ++r) {
#pragma unroll
    for (int off = 1; off < 16; off <<= 1) {
      accth[r] += __shfl_xor(accth[r], off, 32);
      accz[r]  += __shfl_xor(accz[r],  off, 32);
    }
  }
  if (m == 0) {  // lane 0 -> rows t0..t0+7, lane 16 -> rows t0+8..t0+15
    const int base = t0 + 8 * kh;
#pragma unroll
    for (int r = 0; r < 8; ++r) {
      s_theta[base + r] = accth[r];
      s_z[base + r]     = accz[r];
    }
  }
}

// h_t = a*h_{t-1} + (1-a)*s_t over T=32768, 2 channels. Chunked scan:
// per-thread serial scan of 32 elems, serial carry pass, re-apply.
__global__ void __launch_bounds__(1024) onepole_scan(
    const float* __restrict__ s_theta,
    const float* __restrict__ s_z,
    const float* __restrict__ b_theta,  // [1]
    const float* __restrict__ b_z,      // [1]
    float* __restrict__ out)            // [2*T] = theta ++ z
{
  constexpr float ALPHA   = 0.88249690258459546f;   // exp(-1/8)
  constexpr float ONEMA   = 0.11750309741540454f;   // 1 - alpha
  constexpr float ALPHA32 = 0.018315638888734179f;  // alpha^32 = exp(-4)

  __shared__ float Lth[1024], Lz[1024], Cth[1024], Cz[1024];
  const int tid = (int)threadIdx.x;
  const int s0  = tid * 32;

  float pth = 0.0f, pz = 0.0f;
  for (int i = 0; i < 32; ++i) {
    pth = fmaf(ALPHA, pth, ONEMA * s_theta[s0 + i]);
    pz  = fmaf(ALPHA, pz,  ONEMA * s_z[s0 + i]);
  }
  Lth[tid] = pth; Lz[tid] = pz;
  __syncthreads();

  if (tid == 0) {  // exclusive carry scan over 1024 chunks (trivial cost)
    float cth = 0.0f, cz = 0.0f;
    for (int c = 0; c < 1024; ++c) {
      Cth[c] = cth; Cz[c] = cz;
      cth = fmaf(ALPHA32, cth, Lth[c]);
      cz  = fmaf(ALPHA32, cz,  Lz[c]);
    }
  }
  __syncthreads();

  float hth = Cth[tid], hz = Cz[tid];
  const float bth = b_theta[0], bz = b_z[0];
  for (int i = 0; i < 32; ++i) {
    hth = fmaf(ALPHA, hth, ONEMA * s_theta[s0 + i]);
    hz  = fmaf(ALPHA, hz,  ONEMA * s_z[s0 + i]);
    out[s0 + i]         = hth + bth;
    out[T_LEN + s0 + i] = hz + bz;
  }
}

extern "C" void kernel_launch(void* const* d_in, const int* in_sizes, int n_in,
                              void* d_out, int out_size, void* d_ws, size_t ws_size,
                              hipStream_t stream) {
  const float* beliefs = (const float*)d_in[0];
  const float* W_in    = (const float*)d_in[1];
  const float* b_in    = (const float*)d_in[2];
  const float* W_theta = (const float*)d_in[3];
  const float* b_theta = (const float*)d_in[4];
  const float* W_z     = (const float*)d_in[5];
  const float* b_z     = (const float*)d_in[6];

  float* out     = (float*)d_out;
  float* s_theta = (float*)d_ws;
  float* s_z     = s_theta + T_LEN;

  // 256 WGs x 8 waves x 16 rows = 32768 timesteps.
  onepole_gemm<<<dim3(T_LEN / 128), dim3(256), 0, stream>>>(
      beliefs, W_in, b_in, W_theta, W_z, s_theta, s_z);
  onepole_scan<<<dim3(1), dim3(1024), 0, stream>>>(
      s_theta, s_z, b_theta, b_z, out);
}